// RegularizedInfoNCE_6803228197236
// MI455X (gfx1250) — compile-verified
//
#include <hip/hip_runtime.h>
#include <hip/hip_bf16.h>
#include <cmath>

// Problem constants (from the reference)
#define N_ROWS 8192
#define DIM    512
#define INV_T  25.0f   // 1/0.04
#define SHIFT  25.0f   // fixed log-sum-exp shift; logits/T <= 25 for unit vectors

// GEMM tiling
#define BM     64      // rows per workgroup strip
#define BLOCK  128     // 4 waves (wave32) -> 4 x 16 rows
#define LDA    516     // LDS row pitch (pad 4 floats -> bank stride 4, conflict-free)

typedef __attribute__((ext_vector_type(2))) float v2f;
typedef __attribute__((ext_vector_type(8))) float v8f;

// ---------------------------------------------------------------------------
// L2-normalize rows of z1 and z2 (wave per row). row in [0, 2N)
// ---------------------------------------------------------------------------
__global__ __launch_bounds__(BLOCK) void normalize_rows(const float* __restrict__ z1,
                                                        const float* __restrict__ z2,
                                                        float* __restrict__ z1n,
                                                        float* __restrict__ z2n) {
  const int wave = threadIdx.x >> 5;
  const int lane = threadIdx.x & 31;
  const int row  = blockIdx.x * (BLOCK / 32) + wave;   // 0 .. 2N-1
  const float* src;
  float*       dst;
  if (row < N_ROWS) { src = z1 + (size_t)row * DIM;            dst = z1n + (size_t)row * DIM; }
  else              { src = z2 + (size_t)(row - N_ROWS) * DIM; dst = z2n + (size_t)(row - N_ROWS) * DIM; }
  float v[DIM / 32];
  float s = 0.f;
#pragma unroll
  for (int i = 0; i < DIM / 32; ++i) { float x = src[lane + 32 * i]; v[i] = x; s += x * x; }
#pragma unroll
  for (int m = 1; m < 32; m <<= 1) s += __shfl_xor(s, m, 32);
  const float inv = 1.f / fmaxf(sqrtf(s), 1e-12f);
#pragma unroll
  for (int i = 0; i < DIM / 32; ++i) dst[lane + 32 * i] = v[i] * inv;
}

// ---------------------------------------------------------------------------
// diag[i] = dot(z1n[i], z2n[i])  (wave per row)
// ---------------------------------------------------------------------------
__global__ __launch_bounds__(BLOCK) void diag_dot(const float* __restrict__ a,
                                                  const float* __restrict__ b,
                                                  float* __restrict__ diag) {
  const int wave = threadIdx.x >> 5;
  const int lane = threadIdx.x & 31;
  const int row  = blockIdx.x * (BLOCK / 32) + wave;
  const float* pa = a + (size_t)row * DIM;
  const float* pb = b + (size_t)row * DIM;
  float s = 0.f;
#pragma unroll
  for (int i = 0; i < DIM / 32; ++i) s += pa[lane + 32 * i] * pb[lane + 32 * i];
#pragma unroll
  for (int m = 1; m < 32; m <<= 1) s += __shfl_xor(s, m, 32);
  if (lane == 0) diag[row] = s;
}

__global__ void zero_f32(float* __restrict__ p, int n) {
  int i = blockIdx.x * blockDim.x + threadIdx.x;
  if (i < n) p[i] = 0.f;
}

// ---------------------------------------------------------------------------
// Gram strip kernel using V_WMMA_F32_16X16X4_F32 (exact fp32 MACs).
// Each block: 64-row strip of A staged in LDS; 4 waves, 16 rows each, sweep
// all N columns of B producing 16x16 tiles of C = A * B^T.
// MODE 0 (InfoNCE): rowOut[i] = sum_j exp(C_ij*INV_T - SHIFT)  (direct store)
//                   colOut[j] += column partial sums (LDS accum -> global atomic)
// MODE 1 (KoLeo)  : rowOut[i] = max_{j != i} C_ij               (direct store)
// ---------------------------------------------------------------------------
template <int MODE>
__global__ __launch_bounds__(BLOCK) void gram_strip(const float* __restrict__ A,
                                                    const float* __restrict__ B,
                                                    float* __restrict__ rowOut,
                                                    float* __restrict__ colOut) {
  __shared__ float As[BM * LDA];                        // 64 x 516 f32 = 132 KB
  __shared__ float colLds[(MODE == 0) ? N_ROWS : 16];   // 32 KB column accumulators (MODE 0)

  const int tid      = threadIdx.x;
  const int rowBlock = blockIdx.x * BM;

  // Stage the A strip into LDS.
  for (int idx = tid; idx < BM * DIM; idx += BLOCK) {
    const int r = idx >> 9;          // / DIM
    const int k = idx & (DIM - 1);   // % DIM
    As[r * LDA + k] = A[(size_t)(rowBlock + r) * DIM + k];
  }
  if (MODE == 0) {
    for (int j = tid; j < N_ROWS; j += BLOCK) colLds[j] = 0.f;
  }
  __syncthreads();

  const int wave = tid >> 5;         // 0..3
  const int lane = tid & 31;
  const int lmod = lane & 15;        // N index within tile / A row within tile
  const int half = lane >> 4;        // K sub-pair selector per WMMA layout
  const int rowBand = rowBlock + wave * 16;

  const float* AsRow = &As[(wave * 16 + lmod) * LDA];

  float racc[8];
#pragma unroll
  for (int r = 0; r < 8; ++r) racc[r] = (MODE == 0) ? 0.f : -2.f;  // dots >= -1

  for (int colBase = 0; colBase < N_ROWS; colBase += 16) {
    const int col = colBase + lmod;
    const float2* Bp = (const float2*)(B + (size_t)col * DIM);

    v8f c = {};
#pragma unroll 8
    for (int kc = 0; kc < DIM; kc += 4) {
      const float2 af = *(const float2*)(AsRow + kc + 2 * half);  // A: M=lmod, K=kc+2*half..+1
      const float2 bf = Bp[(kc >> 1) + half];                     // B: N=lmod, K=kc+2*half..+1
      v2f a; a.x = af.x; a.y = af.y;
      v2f b; b.x = bf.x; b.y = bf.y;
      // D = A(16x4) * B(4x16) + C, fp32
      c = __builtin_amdgcn_wmma_f32_16x16x4_f32(false, a, false, b, (short)0, c,
                                                false, false);
    }

    if (MODE == 0) {
      float csum = 0.f;
#pragma unroll
      for (int r = 0; r < 8; ++r) {
        const float e = __expf(c[r] * INV_T - SHIFT);
        racc[r] += e;   // row accumulators: lane's rows are fixed all loop long
        csum += e;
      }
      // combine the two half-wave contributions for this column, one LDS atomic
      csum += __shfl_xor(csum, 16, 32);
      if (half == 0) atomicAdd(&colLds[col], csum);
    } else {
#pragma unroll
      for (int r = 0; r < 8; ++r) {
        const int row = rowBand + r + 8 * half;   // C layout: lanes16-31 hold M=8..15
        const float v = (row == col) ? -2.f : c[r];
        racc[r] = fmaxf(racc[r], v);
      }
    }
  }

  // Row results: reduce across the 16 lanes of each half-wave (xor masks < 16
  // stay inside the group), then one lane per half stores its 8 rows.
#pragma unroll
  for (int r = 0; r < 8; ++r) {
    float v = racc[r];
#pragma unroll
    for (int m = 1; m < 16; m <<= 1) {
      const float o = __shfl_xor(v, m, 32);
      v = (MODE == 0) ? (v + o) : fmaxf(v, o);
    }
    if (lmod == 0) rowOut[rowBand + r + 8 * half] = v;
  }

  if (MODE == 0) {
    __syncthreads();
    for (int j = tid; j < N_ROWS; j += BLOCK) atomicAdd(&colOut[j], colLds[j]);
  }
}

// ---------------------------------------------------------------------------
// Final scalar assembly (single block).
// ---------------------------------------------------------------------------
__global__ __launch_bounds__(256) void finalize_loss(const float* __restrict__ diag,
                                                     const float* __restrict__ rowsumS,
                                                     const float* __restrict__ colsumS,
                                                     const float* __restrict__ maxG1,
                                                     const float* __restrict__ maxG2,
                                                     float* __restrict__ out) {
  __shared__ float red[4][256];
  float a12 = 0.f, a21 = 0.f, k1 = 0.f, k2 = 0.f;
  for (int i = threadIdx.x; i < N_ROWS; i += 256) {
    const float dT = diag[i] * INV_T;
    a12 += logf(rowsumS[i]) + SHIFT - dT;   // -logp_ii (rows of S)
    a21 += logf(colsumS[i]) + SHIFT - dT;   // -logp_ii (cols of S)
    const float d1 = sqrtf(fmaxf(2.f - 2.f * maxG1[i], 0.f));
    const float d2 = sqrtf(fmaxf(2.f - 2.f * maxG2[i], 0.f));
    k1 -= logf(d1 + 1e-9f);
    k2 -= logf(d2 + 1e-9f);
  }
  red[0][threadIdx.x] = a12;
  red[1][threadIdx.x] = a21;
  red[2][threadIdx.x] = k1;
  red[3][threadIdx.x] = k2;
  __syncthreads();
  for (int s = 128; s > 0; s >>= 1) {
    if (threadIdx.x < s) {
#pragma unroll
      for (int q = 0; q < 4; ++q) red[q][threadIdx.x] += red[q][threadIdx.x + s];
    }
    __syncthreads();
  }
  if (threadIdx.x == 0) {
    const float invN = 1.0f / (float)N_ROWS;
    const float l12 = red[0][0] * invN;
    const float l21 = red[1][0] * invN;
    const float kk1 = red[2][0] * invN;
    const float kk2 = red[3][0] * invN;
    out[0] = 0.5f * (l12 + l21) + 0.1f * 0.5f * (kk1 + kk2);
  }
}

// ---------------------------------------------------------------------------
extern "C" void kernel_launch(void* const* d_in, const int* in_sizes, int n_in,
                              void* d_out, int out_size, void* d_ws, size_t ws_size,
                              hipStream_t stream) {
  const float* z1 = (const float*)d_in[0];
  const float* z2 = (const float*)d_in[1];
  float* out = (float*)d_out;

  float* ws = (float*)d_ws;
  const size_t ND = (size_t)N_ROWS * DIM;
  float* z1n     = ws;
  float* z2n     = ws + ND;
  float* diag    = ws + 2 * ND;
  float* rowsumS = diag + N_ROWS;
  float* colsumS = rowsumS + N_ROWS;
  float* maxG1   = colsumS + N_ROWS;
  float* maxG2   = maxG1 + N_ROWS;

  const int rowsPerBlk = BLOCK / 32;
  normalize_rows<<<(2 * N_ROWS) / rowsPerBlk, BLOCK, 0, stream>>>(z1, z2, z1n, z2n);
  diag_dot<<<N_ROWS / rowsPerBlk, BLOCK, 0, stream>>>(z1n, z2n, diag);
  zero_f32<<<(N_ROWS + 255) / 256, 256, 0, stream>>>(colsumS, N_ROWS);

  const int nStrips = N_ROWS / BM;
  gram_strip<0><<<nStrips, BLOCK, 0, stream>>>(z1n, z2n, rowsumS, colsumS);
  gram_strip<1><<<nStrips, BLOCK, 0, stream>>>(z1n, z1n, maxG1, nullptr);
  gram_strip<1><<<nStrips, BLOCK, 0, stream>>>(z2n, z2n, maxG2, nullptr);

  finalize_loss<<<1, 256, 0, stream>>>(diag, rowsumS, colsumS, maxG1, maxG2, out);
}